// GaussianRasterizer_61040075210908
// MI455X (gfx1250) — compile-verified
//
#include <hip/hip_runtime.h>
#include <math.h>

#define N_G 2048
#define IMG_H 128
#define IMG_W 128
#define TANFOV 0.5f
#define NEARP 0.2f
#define ALPHA_MIN (1.0f/255.0f)
#define REC_STRIDE 12
#define STAGE_G 64
#define STAGE_FLOATS (STAGE_G * REC_STRIDE)   /* 768 dwords = 3 KB per stage */
#define NSTAGE (N_G / STAGE_G)                /* 32 */

typedef __attribute__((ext_vector_type(2))) float v2f;
typedef __attribute__((ext_vector_type(8))) float v8f;
typedef __attribute__((ext_vector_type(4))) unsigned int u32x4;
typedef __attribute__((ext_vector_type(8))) unsigned int u32x8;
typedef __attribute__((address_space(3))) float lds_float_t;

// record layout: 0 px, 1 py, 2 conA, 3 conB, 4 conC, 5 op, 6 r, 7 g, 8 b, 9..11 pad

__global__ __launch_bounds__(256) void preprocess_kernel(
    const float* __restrict__ means3D, const float* __restrict__ colors,
    const float* __restrict__ opac, const float* __restrict__ cov3D,
    float* __restrict__ rec, float* __restrict__ zkey, int* __restrict__ radii_out)
{
  int i = blockIdx.x * blockDim.x + threadIdx.x;
  if (i >= N_G) return;
  const float fx = IMG_W / (2.f * TANFOV), fy = IMG_H / (2.f * TANFOV);
  float x = means3D[3*i+0], y = means3D[3*i+1], z = means3D[3*i+2];
  float iz = 1.f / z;
  float px = fx * x * iz + 0.5f * IMG_W;
  float py = fy * y * iz + 0.5f * IMG_H;
  const float lim = 1.3f * TANFOV;
  float cxn = fminf(fmaxf(x * iz, -lim), lim);
  float cyn = fminf(fmaxf(y * iz, -lim), lim);
  float s00 = cov3D[6*i+0], s01 = cov3D[6*i+1], s02 = cov3D[6*i+2];
  float s11 = cov3D[6*i+3], s12 = cov3D[6*i+4], s22 = cov3D[6*i+5];
  // J = [[fx*iz, 0, -fx*cxn*iz], [0, fy*iz, -fy*cyn*iz]]  (tx*iz^2 == cxn*iz)
  float j00 = fx * iz, j02 = -fx * cxn * iz;
  float j11 = fy * iz, j12 = -fy * cyn * iz;
  float m00 = j00*s00 + j02*s02;
  float m01 = j00*s01 + j02*s12;
  float m02 = j00*s02 + j02*s22;
  float m11 = j11*s11 + j12*s12;
  float m12 = j11*s12 + j12*s22;
  float a = m00*j00 + m02*j02 + 0.3f;
  float b = m01*j11 + m02*j12;
  float c = m11*j11 + m12*j12 + 0.3f;
  float det = a*c - b*b;
  float invdet = 1.f / det;
  float conA = c * invdet, conB = -b * invdet, conC = a * invdet;
  float mid = 0.5f * (a + c);
  float lam1 = mid + sqrtf(fmaxf(0.1f, mid*mid - det));
  bool valid = (z > NEARP) && (det > 0.f);
  radii_out[i] = valid ? (int)ceilf(3.f * sqrtf(lam1)) : 0;
  float* r = rec + (size_t)i * REC_STRIDE;
  r[0] = px; r[1] = py; r[2] = conA; r[3] = conB; r[4] = conC;
  r[5] = valid ? opac[i] : 0.f;
  r[6] = colors[3*i+0]; r[7] = colors[3*i+1]; r[8] = colors[3*i+2];
  r[9] = 0.f; r[10] = 0.f; r[11] = 0.f;
  zkey[i] = z;
}

// Single-workgroup bitonic argsort over 2048 z keys (1024 threads = N/2 compare pairs)
__global__ __launch_bounds__(1024) void sort_kernel(const float* __restrict__ zkey,
                                                    int* __restrict__ idx_out)
{
  __shared__ float k[N_G];
  __shared__ int v[N_G];
  int t = threadIdx.x;
  for (int i = t; i < N_G; i += 1024) { k[i] = zkey[i]; v[i] = i; }
  __syncthreads();
  for (int size = 2; size <= N_G; size <<= 1) {
    for (int stride = size >> 1; stride > 0; stride >>= 1) {
      int i = (t / stride) * (stride * 2) + (t % stride);
      int j = i + stride;
      bool asc = ((i & size) == 0);
      float ki = k[i], kj = k[j];
      if ((ki > kj) == asc) {
        k[i] = kj; k[j] = ki;
        int tv = v[i]; v[i] = v[j]; v[j] = tv;
      }
      __syncthreads();
    }
  }
  for (int i = t; i < N_G; i += 1024) idx_out[i] = v[i];
}

__global__ __launch_bounds__(256) void gather_kernel(const float* __restrict__ rec,
                                                     const int* __restrict__ idx,
                                                     float* __restrict__ srec)
{
  int i = blockIdx.x * blockDim.x + threadIdx.x;
  if (i >= N_G) return;
  int j = idx[i];
  const float* s = rec + (size_t)j * REC_STRIDE;
  float* d = srec + (size_t)i * REC_STRIDE;
  #pragma unroll
  for (int q = 0; q < REC_STRIDE; ++q) d[q] = s[q];
}

// Issue one TDM tensor load: 768 contiguous dwords (one stage of sorted records)
// from global `src` into LDS byte address `lds_addr`. D# per CDNA5 ISA.
__device__ __forceinline__ void tdm_load_stage(const float* src, unsigned lds_addr)
{
  unsigned long long ga = (unsigned long long)src;
  u32x4 g0;
  g0[0] = 1u;                                        // count=1, user mode
  g0[1] = lds_addr;                                  // lds_addr (bytes)
  g0[2] = (unsigned)(ga & 0xFFFFFFFFull);            // global_addr[31:0]
  g0[3] = (unsigned)((ga >> 32) & 0x01FFFFFFull)     // global_addr[56:32]
        | (2u << 30);                                // type=2 (image)
  u32x8 g1;
  g1[0] = 0x00020000u;                               // wg_mask=0, data_size=2 (4B)
  g1[1] = ((unsigned)STAGE_FLOATS & 0xFFFFu) << 16;  // tensor_dim0[15:0]
  g1[2] = ((unsigned)STAGE_FLOATS >> 16)             // tensor_dim0[31:16]
        | (1u << 16);                                // tensor_dim1 = 1
  g1[3] = ((unsigned)STAGE_FLOATS) << 16;            // tile_dim0 = 768
  g1[4] = 1u;                                        // tile_dim1 = 1, tile_dim2 = 0
  g1[5] = (unsigned)STAGE_FLOATS;                    // tensor_dim0_stride
  g1[6] = 0u;
  g1[7] = 0u;
  asm volatile("tensor_load_to_lds %0, %1" :: "s"(g0), "s"(g1) : "memory");
}

// 256 blocks x 128 threads; each wave (wave32) owns a 4x4 pixel tile.
// Stages of 64 gaussians are DMA'd to LDS by the Tensor Data Mover (double-
// buffered, overlapping the next stage's DMA with this stage's WMMA+compositing).
// Per 16-gaussian chunk: power tile via 2x V_WMMA_F32_16X16X4_F32 (K padded 6->8),
// then ordered alpha compositing (8 gaussians/lane, shfl-merge across halves).
__global__ __launch_bounds__(128) void raster_kernel(const float* __restrict__ srec,
                                                     const float* __restrict__ bg,
                                                     float* __restrict__ img)
{
  __shared__ __align__(16) float lrec[2][STAGE_FLOATS];
  int tid = threadIdx.x;
  int wave = tid >> 5;
  int lane = tid & 31;
  int p = lane & 15;       // pixel within 4x4 wave tile (also A-matrix row M)
  int h = lane >> 4;       // half select (K pair / depth half)
  int lx = p & 3, ly = p >> 2;
  int bx = blockIdx.x & 15, by = blockIdx.x >> 4;
  int tileX = bx * 8 + (wave & 1) * 4;
  int tileY = by * 8 + (wave >> 1) * 4;
  int X = tileX + lx, Y = tileY + ly;

  unsigned ldsA0 = (unsigned)(unsigned long long)(lds_float_t*)&lrec[0][0];
  unsigned ldsA1 = (unsigned)(unsigned long long)(lds_float_t*)&lrec[1][0];

  // B matrix: pixel basis [1, x, y, x^2, y^2, xy, 0, 0] in tile-local coords.
  // Lane supplies K pair (2h, 2h+1) in B0 and (4+2h, 5+2h) in B1.
  float flx = (float)lx, fly = (float)ly;
  v2f B0, B1;
  B0.x = h ? fly       : 1.f;        // b2 : b0
  B0.y = h ? flx * fly : flx;        // hmm: b3 is x^2 -- fix below
  // basis order: [1, x, y, x*x, y*y, x*y, 0, 0]
  B0.x = h ? fly       : 1.f;        // K2=y   : K0=1
  B0.y = h ? flx * flx : flx;        // K3=x^2 : K1=x
  B1.x = h ? 0.f       : fly * fly;  // K6=0   : K4=y^2
  B1.y = h ? 0.f       : flx * fly;  // K7=0   : K5=x*y

  float T = 1.f, Cr = 0.f, Cg = 0.f, Cb = 0.f;

  // Prologue: DMA stage 0 into buffer 0.
  if (wave == 0)
    tdm_load_stage(srec, ldsA0);

  for (int stage = 0; stage < NSTAGE; ++stage) {
    if (wave == 0)
      __builtin_amdgcn_s_wait_tensorcnt(0);   // current stage's DMA complete
    __syncthreads();                          // all waves see buffer, and all
                                              // finished with the other buffer
    if (wave == 0 && stage + 1 < NSTAGE)
      tdm_load_stage(srec + (size_t)(stage + 1) * STAGE_FLOATS,
                     ((stage + 1) & 1) ? ldsA1 : ldsA0);

    const float* cur = &lrec[stage & 1][0];

    for (int sub = 0; sub < STAGE_G / 16; ++sub) {
      int base = sub * 16;
      // Build A row for gaussian M = p; lane supplies K pair selected by h.
      const float* g = &cur[(base + p) * REC_STRIDE];
      float gpx = g[0], gpy = g[1], cA = g[2], cB = g[3], cC = g[4];
      float u = gpx - (float)tileX, v = gpy - (float)tileY;
      float k0 = -0.5f * (cA*u*u + cC*v*v) - cB*u*v;   // const
      float k1 = cA*u + cB*v;                          // * x
      float k2 = cC*v + cB*u;                          // * y
      float k3 = -0.5f * cA;                           // * x^2
      float k4 = -0.5f * cC;                           // * y^2
      float k5 = -cB;                                  // * x*y
      v2f A0, A1;
      A0.x = h ? k2  : k0;
      A0.y = h ? k3  : k1;
      A1.x = h ? 0.f : k4;
      A1.y = h ? 0.f : k5;

      v8f acc = {};
      acc = __builtin_amdgcn_wmma_f32_16x16x4_f32(false, A0, false, B0, (short)0, acc, false, false);
      acc = __builtin_amdgcn_wmma_f32_16x16x4_f32(false, A1, false, B1, (short)0, acc, false, false);

      // Sequential compositing over this lane's 8 gaussians (D: M = r + 8*h)
      float tl = 1.f, cr = 0.f, cg = 0.f, cb = 0.f;
      #pragma unroll
      for (int r = 0; r < 8; ++r) {
        int gi = base + 8*h + r;
        const float* gr = &cur[gi * REC_STRIDE];
        float pw = acc[r];
        float alpha = fminf(0.99f, gr[5] * __expf(pw));
        if (pw > 0.f || alpha < ALPHA_MIN) alpha = 0.f;
        float w = tl * alpha;
        cr += w * gr[6]; cg += w * gr[7]; cb += w * gr[8];
        tl *= (1.f - alpha);
      }
      // Merge depth halves in order via shfl_xor(16)
      float tp = __shfl_xor(tl, 16);
      float pr = __shfl_xor(cr, 16);
      float pg = __shfl_xor(cg, 16);
      float pb = __shfl_xor(cb, 16);
      float chr, chg, chb;
      if (h == 0) { chr = cr + tl*pr; chg = cg + tl*pg; chb = cb + tl*pb; }
      else        { chr = pr + tp*cr; chg = pg + tp*cg; chb = pb + tp*cb; }
      float chT = tl * tp;
      Cr += T * chr; Cg += T * chg; Cb += T * chb;
      T *= chT;
    }
  }

  if (h == 0) {
    int pix = Y * IMG_W + X;
    img[0*IMG_H*IMG_W + pix] = Cr + T * bg[0];
    img[1*IMG_H*IMG_W + pix] = Cg + T * bg[1];
    img[2*IMG_H*IMG_W + pix] = Cb + T * bg[2];
  }
}

extern "C" void kernel_launch(void* const* d_in, const int* in_sizes, int n_in,
                              void* d_out, int out_size, void* d_ws, size_t ws_size,
                              hipStream_t stream)
{
  const float* means3D = (const float*)d_in[0];
  const float* colors  = (const float*)d_in[1];
  const float* opac    = (const float*)d_in[2];
  const float* cov3D   = (const float*)d_in[3];
  const float* bg      = (const float*)d_in[4];

  float* img   = (float*)d_out;
  int*   radii = (int*)d_out + 3 * IMG_H * IMG_W;

  float* ws   = (float*)d_ws;
  float* rec  = ws;                                   // N_G*12 floats
  float* zkey = ws + N_G * REC_STRIDE;                // N_G floats
  int*   sidx = (int*)(ws + N_G * REC_STRIDE + N_G);  // N_G ints
  float* srec = ws + N_G * REC_STRIDE + 2 * N_G;      // N_G*12 floats (sorted)

  preprocess_kernel<<<(N_G + 255) / 256, 256, 0, stream>>>(means3D, colors, opac, cov3D,
                                                           rec, zkey, radii);
  sort_kernel<<<1, 1024, 0, stream>>>(zkey, sidx);
  gather_kernel<<<(N_G + 255) / 256, 256, 0, stream>>>(rec, sidx, srec);
  raster_kernel<<<256, 128, 0, stream>>>(srec, bg, img);
}